// GATUnit_34067680592302
// MI455X (gfx1250) — compile-verified
//
#include <hip/hip_runtime.h>
#include <cstdint>

typedef float v2f __attribute__((ext_vector_type(2)));
typedef float v8f __attribute__((ext_vector_type(8)));

#define NN 50000
#define EE 800000
#define ETOT (EE + NN)          // 850000 edges incl. self loops
#define NEG_SLOPE 0.2f

// ---- order-preserving float<->uint mapping for atomic segment-max ----
__device__ __forceinline__ unsigned f2ord(float f) {
  unsigned u = __float_as_uint(f);
  return (u & 0x80000000u) ? ~u : (u | 0x80000000u);
}
__device__ __forceinline__ float ord2f(unsigned u) {
  return (u & 0x80000000u) ? __uint_as_float(u & 0x7fffffffu) : __uint_as_float(~u);
}

__global__ void zero_f32(float* p, int n) {
  int i = blockIdx.x * blockDim.x + threadIdx.x;
  if (i < n) p[i] = 0.0f;
}
__global__ void zero_u32(unsigned* p, int n) {
  int i = blockIdx.x * blockDim.x + threadIdx.x;
  if (i < n) p[i] = 0u;   // floor value for the ordered-uint max
}

// y[n4 float4] = broadcast bias rows (128 floats = 32 float4 per row).
// Aggregation then atomically accumulates on top -> bias folded for free.
__global__ void init_bias(float4* __restrict__ y, const float4* __restrict__ b, int n4) {
  int i = blockIdx.x * blockDim.x + threadIdx.x;
  if (i < n4) y[i] = b[i & 31];
}

// ---------------------------------------------------------------------
// Y[nrows,128] = X[nrows,128] @ W[128,128], fp32, V_WMMA_F32_16X16X4_F32.
// grid.x = nrows/16 row tiles; block = 256 threads = 8 waves, wave w owns
// output columns [16w,16w+16). K looped 128/4 = 32 WMMAs per wave.
// A 16x4 layout: lane(0..15)=row M, half-wave selects K pair, VGPR = K lsb.
// B 4x16 layout: lane(0..15)=col N, half-wave selects K pair, VGPR = K lsb.
// C/D 16x16: VGPR r -> row r (lanes 0..15) / row r+8 (lanes 16..31).
// ---------------------------------------------------------------------
__global__ void gemm128_wmma(const float* __restrict__ X,
                             const float* __restrict__ W,
                             float* __restrict__ Y) {
  const int lane = threadIdx.x & 31;
  const int wave = threadIdx.x >> 5;
  const int row0 = blockIdx.x << 4;
  const int col0 = wave << 4;
  const int m  = lane & 15;            // row (A) / col (B) within tile
  const int kh = (lane >> 4) << 1;     // K sub-offset: 0 or 2

  v8f acc = {};
  const float* __restrict__ xrow = X + (size_t)(row0 + m) * 128;
  const float* __restrict__ wcol = W + col0 + m;

  #pragma unroll 8
  for (int k0 = 0; k0 < 128; k0 += 4) {
    v2f a, b;
    a.x = xrow[k0 + kh];
    a.y = xrow[k0 + kh + 1];
    b.x = wcol[(size_t)(k0 + kh) * 128];
    b.y = wcol[(size_t)(k0 + kh + 1) * 128];
    acc = __builtin_amdgcn_wmma_f32_16x16x4_f32(false, a, false, b,
                                                (short)0, acc, false, false);
  }

  const int mb = (lane >> 4) << 3;     // 0 or 8
  float* yo = Y + ((size_t)row0 + mb) * 128 + col0 + m;
  #pragma unroll
  for (int r = 0; r < 8; ++r) yo[(size_t)r * 128] = acc[r];
}

// ---------------------------------------------------------------------
// Edge kernels: one wave32 per edge; lane handles 4 consecutive channels
// (float4 = global_load_b128). H=2 -> 16-lane groups per head; H=1 -> 32.
// ---------------------------------------------------------------------
__device__ __forceinline__ void edge_ends(int e, const int* __restrict__ ei,
                                          int& src, int& dst) {
  if (e < EE) { src = ei[e]; dst = ei[EE + e]; }
  else        { src = dst = e - EE; }
}

template <int H>
__global__ void edge_score(const float* __restrict__ xl,
                           const float* __restrict__ xr,
                           const int* __restrict__ ei,
                           const float* __restrict__ att,   // flattened [128]
                           float* __restrict__ s,
                           unsigned* __restrict__ smax) {
  const int lane = threadIdx.x & 31;
  const int e = blockIdx.x * (blockDim.x >> 5) + (threadIdx.x >> 5);
  if (e >= ETOT) return;
  int src, dst;
  edge_ends(e, ei, src, dst);

  const float4 a = *(const float4*)(xl + (size_t)src * 128 + lane * 4);
  const float4 b = *(const float4*)(xr + (size_t)dst * 128 + lane * 4);
  const float4 w = *(const float4*)(att + lane * 4);

  float t0 = a.x + b.x, t1 = a.y + b.y, t2 = a.z + b.z, t3 = a.w + b.w;
  t0 = (t0 > 0.f) ? t0 : NEG_SLOPE * t0;
  t1 = (t1 > 0.f) ? t1 : NEG_SLOPE * t1;
  t2 = (t2 > 0.f) ? t2 : NEG_SLOPE * t2;
  t3 = (t3 > 0.f) ? t3 : NEG_SLOPE * t3;
  float p = t0 * w.x + t1 * w.y + t2 * w.z + t3 * w.w;

  const int grp = 32 / H;
  #pragma unroll
  for (int off = 1; off < grp; off <<= 1) p += __shfl_xor(p, off, 32);

  if ((lane & (grp - 1)) == 0) {
    const int h = lane / grp;
    s[(size_t)e * H + h] = p;
    atomicMax(&smax[(size_t)dst * H + h], f2ord(p));
  }
}

template <int H>
__global__ void edge_exp(const int* __restrict__ ei,
                         float* __restrict__ s,
                         const unsigned* __restrict__ smax,
                         float* __restrict__ denom) {
  const int e = blockIdx.x * blockDim.x + threadIdx.x;   // one thread per edge
  if (e >= ETOT) return;
  int src, dst;
  edge_ends(e, ei, src, dst);
  (void)src;
  #pragma unroll
  for (int h = 0; h < H; ++h) {
    const float m  = ord2f(smax[(size_t)dst * H + h]);
    const float ex = __expf(s[(size_t)e * H + h] - m);
    s[(size_t)e * H + h] = ex;                 // overwrite score with exp
    atomicAdd(&denom[(size_t)dst * H + h], ex);
  }
}

template <int H>
__global__ void edge_aggr(const float* __restrict__ xl,
                          const int* __restrict__ ei,
                          const float* __restrict__ s,      // holds exp()
                          const float* __restrict__ denom,
                          float* __restrict__ out) {
  const int lane = threadIdx.x & 31;
  const int e = blockIdx.x * (blockDim.x >> 5) + (threadIdx.x >> 5);
  if (e >= ETOT) return;
  int src, dst;
  edge_ends(e, ei, src, dst);

  const int h = (H == 2) ? (lane >> 4) : 0;
  const float alpha = s[(size_t)e * H + h] / denom[(size_t)dst * H + h];

  const float4 a = *(const float4*)(xl + (size_t)src * 128 + lane * 4);
  float* o = out + (size_t)dst * 128 + lane * 4;
  atomicAdd(o + 0, alpha * a.x);
  atomicAdd(o + 1, alpha * a.y);
  atomicAdd(o + 2, alpha * a.z);
  atomicAdd(o + 3, alpha * a.w);
}

// ---------------------------------------------------------------------
extern "C" void kernel_launch(void* const* d_in, const int* in_sizes, int n_in,
                              void* d_out, int out_size, void* d_ws, size_t ws_size,
                              hipStream_t stream) {
  (void)in_sizes; (void)n_in; (void)out_size; (void)ws_size;

  const float* x    = (const float*)d_in[0];
  const int*   ei   = (const int*)d_in[1];   // [2,E]: row0 = src, row1 = dst
  // d_in[2] = edge_attr (unused by reference path)
  const float* Wl1  = (const float*)d_in[3];
  const float* Wr1  = (const float*)d_in[4];
  const float* att1 = (const float*)d_in[5];
  const float* b1   = (const float*)d_in[6];
  const float* Wl2  = (const float*)d_in[7];
  const float* Wr2  = (const float*)d_in[8];
  const float* att2 = (const float*)d_in[9];
  const float* b2   = (const float*)d_in[10];
  float* out = (float*)d_out;

  // workspace layout (floats): xl | xr | h | s | denom | smax  (~84.5 MB)
  float*    xl    = (float*)d_ws;
  float*    xr    = xl  + (size_t)NN * 128;
  float*    hbuf  = xr  + (size_t)NN * 128;
  float*    sbuf  = hbuf + (size_t)NN * 128;
  float*    denom = sbuf + (size_t)ETOT * 2;
  unsigned* smax  = (unsigned*)(denom + (size_t)NN * 2);

  const int NF4  = NN * 32;                  // node features as float4
  const dim3 b256(256);
  const int gGemm = NN / 16;                 // 3125 row tiles
  const int gEdge = (ETOT + 7) / 8;          // wave per edge, 8 waves/block
  const int gE1   = (ETOT + 255) / 256;
  const int gNF4  = (NF4 + 255) / 256;

  // ---------------- layer 1: 128 -> 2 heads x 64 ----------------
  gemm128_wmma<<<gGemm, b256, 0, stream>>>(x, Wl1, xl);
  gemm128_wmma<<<gGemm, b256, 0, stream>>>(x, Wr1, xr);
  zero_u32<<<(NN * 2 + 255) / 256, b256, 0, stream>>>(smax, NN * 2);
  zero_f32<<<(NN * 2 + 255) / 256, b256, 0, stream>>>(denom, NN * 2);
  init_bias<<<gNF4, b256, 0, stream>>>((float4*)hbuf, (const float4*)b1, NF4);
  edge_score<2><<<gEdge, b256, 0, stream>>>(xl, xr, ei, att1, sbuf, smax);
  edge_exp<2><<<gE1, b256, 0, stream>>>(ei, sbuf, smax, denom);
  edge_aggr<2><<<gEdge, b256, 0, stream>>>(xl, ei, sbuf, denom, hbuf);

  // ---------------- layer 2: 128 -> 1 head x 128 ----------------
  gemm128_wmma<<<gGemm, b256, 0, stream>>>(hbuf, Wl2, xl);
  gemm128_wmma<<<gGemm, b256, 0, stream>>>(hbuf, Wr2, xr);
  zero_u32<<<(NN + 255) / 256, b256, 0, stream>>>(smax, NN);
  zero_f32<<<(NN + 255) / 256, b256, 0, stream>>>(denom, NN);
  init_bias<<<gNF4, b256, 0, stream>>>((float4*)out, (const float4*)b2, NF4);
  edge_score<1><<<gEdge, b256, 0, stream>>>(xl, xr, ei, att2, sbuf, smax);
  edge_exp<1><<<gE1, b256, 0, stream>>>(ei, sbuf, smax, denom);
  edge_aggr<1><<<gEdge, b256, 0, stream>>>(xl, ei, sbuf, denom, out);
}